// TensorProductReference_62345745268779
// MI455X (gfx1250) — compile-verified
//
#include <hip/hip_runtime.h>
#include <cstdint>
#include <cstddef>
#include <utility>

// ============================================================================
// Compile-time real Wigner-3j (Clebsch-Gordan) tables — constexpr port of the
// reference's _w3j / _real_basis / _real_wigner3j. All coefficients fold to
// immediates in the fully-unrolled device code.
// ============================================================================
namespace cgconst {

constexpr double cfact(int n) {
  double r = 1.0;
  for (int i = 2; i <= n; ++i) r *= double(i);
  return r;
}

constexpr double csqrt(double x) {
  if (x <= 0.0) return 0.0;
  double g = x > 1.0 ? x : 1.0;
  for (int i = 0; i < 100; ++i) g = 0.5 * (g + x / g);
  return g;
}

constexpr double w3j(int j1, int j2, int j3, int m1, int m2, int m3) {
  if (m1 + m2 + m3 != 0) return 0.0;
  int lo = j1 - j2; if (lo < 0) lo = -lo;
  if (j3 < lo || j3 > j1 + j2) return 0.0;
  if (m1 > j1 || m1 < -j1 || m2 > j2 || m2 < -j2 || m3 > j3 || m3 < -j3) return 0.0;
  const double tri = cfact(j1 + j2 - j3) * cfact(j1 - j2 + j3) * cfact(-j1 + j2 + j3)
                     / cfact(j1 + j2 + j3 + 1);
  const double pre = csqrt(tri * cfact(j1 + m1) * cfact(j1 - m1) * cfact(j2 + m2)
                               * cfact(j2 - m2) * cfact(j3 + m3) * cfact(j3 - m3));
  double s = 0.0;
  for (int k = 0; k <= j1 + j2 - j3; ++k) {
    const int d1 = j1 + j2 - j3 - k, d2 = j1 - m1 - k, d3 = j2 + m2 - k;
    const int d4 = j3 - j2 + m1 + k, d5 = j3 - j1 - m2 + k;
    if (d1 < 0 || d2 < 0 || d3 < 0 || d4 < 0 || d5 < 0) continue;
    const double den = cfact(k) * cfact(d1) * cfact(d2) * cfact(d3) * cfact(d4) * cfact(d5);
    s += ((k & 1) ? -1.0 : 1.0) / den;
  }
  int p = j1 - j2 - m3; if (p < 0) p = -p;
  return ((p & 1) ? -1.0 : 1.0) * pre * s;
}

// One row of the complex->real basis matrix Q (at most 2 nonzeros per row).
struct QR { int n; int col[2]; double re[2]; double im[2]; };

constexpr QR qrow(int L, int a) {
  QR r{};
  const int m = a - L;
  const double s2 = 1.0 / csqrt(2.0);
  if (m == 0) {
    r.n = 1; r.col[0] = L; r.re[0] = 1.0; r.im[0] = 0.0;
  } else if (m > 0) {
    r.n = 2;
    r.col[0] = L + m; r.re[0] = ((m & 1) ? -1.0 : 1.0) * s2; r.im[0] = 0.0;
    r.col[1] = L - m; r.re[1] = s2;                          r.im[1] = 0.0;
  } else {
    const int mm = -m;
    r.n = 2;
    r.col[0] = L + m; r.re[0] = 0.0; r.im[0] = s2;
    r.col[1] = L - m; r.re[1] = 0.0; r.im[1] = -(((mm & 1) ? -1.0 : 1.0)) * s2;
  }
  return r;
}

template <int L1, int L2, int L3>
struct CGTab { float w[2 * L1 + 1][2 * L2 + 1][2 * L3 + 1]; };

template <int L1, int L2, int L3>
constexpr CGTab<L1, L2, L3> make_cg() {
  constexpr int N1 = 2 * L1 + 1, N2 = 2 * L2 + 1, N3 = 2 * L3 + 1;
  double Wc[N1][N2][N3]{};
  for (int A = 0; A < N1; ++A)
    for (int B = 0; B < N2; ++B)
      for (int C = 0; C < N3; ++C)
        Wc[A][B][C] = w3j(L1, L2, L3, A - L1, B - L2, C - L3);

  double Tr[N1][N2][N3]{};
  double Ti[N1][N2][N3]{};
  for (int a = 0; a < N1; ++a) {
    const QR qa = qrow(L1, a);
    for (int b = 0; b < N2; ++b) {
      const QR qb = qrow(L2, b);
      for (int c = 0; c < N3; ++c) {
        const QR qc = qrow(L3, c);
        double re = 0.0, im = 0.0;
        for (int i = 0; i < qa.n; ++i)
          for (int j = 0; j < qb.n; ++j)
            for (int k = 0; k < qc.n; ++k) {
              const double w = Wc[qa.col[i]][qb.col[j]][qc.col[k]];
              if (w == 0.0) continue;
              const double r1 = qa.re[i], i1 = qa.im[i];
              const double r2 = qb.re[j], i2 = qb.im[j];
              const double r3 = qc.re[k], i3 = qc.im[k];
              const double pr = r1 * r2 - i1 * i2, pi = r1 * i2 + i1 * r2;
              re += w * (pr * r3 - pi * i3);
              im += w * (pr * i3 + pi * r3);
            }
        Tr[a][b][c] = re;
        Ti[a][b][c] = im;
      }
    }
  }
  double nr = 0.0, ni = 0.0;
  for (int a = 0; a < N1; ++a)
    for (int b = 0; b < N2; ++b)
      for (int c = 0; c < N3; ++c) {
        nr += Tr[a][b][c] * Tr[a][b][c];
        ni += Ti[a][b][c] * Ti[a][b][c];
      }
  const bool useR = (nr >= ni);
  const double scale = csqrt(2.0 * L3 + 1.0) / csqrt(useR ? nr : ni);
  CGTab<L1, L2, L3> t{};
  for (int a = 0; a < N1; ++a)
    for (int b = 0; b < N2; ++b)
      for (int c = 0; c < N3; ++c)
        t.w[a][b][c] = (float)(((useR ? Tr : Ti)[a][b][c]) * scale);
  return t;
}

template <int L1, int L2, int L3>
struct Hold { static constexpr CGTab<L1, L2, L3> tab = make_cg<L1, L2, L3>(); };

template <int L1, int L2, int L3>
constexpr bool any_k(int a, int b) {
  for (int k = 0; k < 2 * L3 + 1; ++k)
    if (Hold<L1, L2, L3>::tab.w[a][b][k] != 0.0f) return true;
  return false;
}

}  // namespace cgconst

// ---------------------------------------------------------------------------
// constexpr-index unroll helper (guarantees compile-time coefficient folding)
// ---------------------------------------------------------------------------
template <class F, int... I>
__device__ __forceinline__ void unroll_impl(F&& f, std::integer_sequence<int, I...>) {
  (f(std::integral_constant<int, I>{}), ...);
}
template <int N, class F>
__device__ __forceinline__ void unroll(F&& f) {
  unroll_impl(static_cast<F&&>(f), std::make_integer_sequence<int, N>{});
}

// One (l1,l2,l3) instruction: acc[k] += w[a][b][k] * x[a] * y[b], then NT store
// to this instruction's exclusive output block at row offset KO.
template <int L1, int L2, int L3, int KO>
__device__ __forceinline__ void do_ins(const float (&xv)[16], const float (&yv)[16],
                                       float* outp) {
  float acc[2 * L3 + 1] = {};
  unroll<2 * L1 + 1>([&](auto AC) {
    constexpr int a = decltype(AC)::value;
    unroll<2 * L2 + 1>([&](auto BC) {
      constexpr int b = decltype(BC)::value;
      if constexpr (cgconst::any_k<L1, L2, L3>(a, b)) {
        const float xy = xv[L1 * L1 + a] * yv[L2 * L2 + b];
        unroll<2 * L3 + 1>([&](auto KC) {
          constexpr int k = decltype(KC)::value;
          constexpr float w = cgconst::Hold<L1, L2, L3>::tab.w[a][b][k];
          if constexpr (w != 0.0f) {
            acc[k] = __builtin_fmaf(w, xy, acc[k]);
          }
        });
      }
    });
  });
  unroll<2 * L3 + 1>([&](auto KC) {
    constexpr int k = decltype(KC)::value;
    __builtin_nontemporal_store(acc[k], outp + (size_t)(KO + k) * 64);
  });
}

// ============================================================================
// Kernel: 256 threads = 8 wave32 = 4 edges (2 channel-halves per edge).
// Each wave stages its own x/y slice (16 rows x 32 ch x 4B x 2 tensors = 4KB)
// into LDS with gfx1250 async-to-LDS b128 ops, waits on ASYNCcnt (wave-private
// staging -> no barrier), then runs the fully-unrolled sparse contraction and
// streams 99 floats/thread out with non-temporal stores.
// ============================================================================
__global__ __launch_bounds__(256) void tp_kernel(const float* __restrict__ x,
                                                 const float* __restrict__ y,
                                                 float* __restrict__ out,
                                                 int n_edges) {
  __shared__ float xs[8][16][32];
  __shared__ float ys[8][16][32];

  const int lane = threadIdx.x & 31;
  const int wave = threadIdx.x >> 5;
  const int half = wave & 1;                     // channel half: 0 -> ch 0..31, 1 -> ch 32..63
  const int edge = blockIdx.x * 4 + (wave >> 1); // wave-uniform
  if (edge >= n_edges) return;

  const unsigned row4 = (unsigned)(lane >> 3);      // 4 rows per b128 op
  const unsigned cof  = (unsigned)((lane & 7) * 4); // 4 floats (16B) per lane
  const size_t gbase = ((size_t)edge * 16) * 64 + (size_t)(half * 32);
  const unsigned ldsx = (unsigned)(uintptr_t)(&xs[0][0][0]) + (unsigned)(wave * 16 * 32 * 4);
  const unsigned ldsy = (unsigned)(uintptr_t)(&ys[0][0][0]) + (unsigned)(wave * 16 * 32 * 4);

#pragma unroll
  for (int i = 0; i < 4; ++i) {
    const unsigned row = (unsigned)(i * 4) + row4;
    const unsigned loff = (row * 32u + cof) * 4u;
    unsigned long long gx =
        (unsigned long long)(uintptr_t)(x + gbase + (size_t)row * 64 + cof);
    unsigned long long gy =
        (unsigned long long)(uintptr_t)(y + gbase + (size_t)row * 64 + cof);
    const unsigned lx = ldsx + loff;
    const unsigned ly = ldsy + loff;
    asm volatile("global_load_async_to_lds_b128 %0, %1, off" ::"v"(lx), "v"(gx) : "memory");
    asm volatile("global_load_async_to_lds_b128 %0, %1, off" ::"v"(ly), "v"(gy) : "memory");
  }
  asm volatile("s_wait_asynccnt 0" ::: "memory");

  float xv[16], yv[16];
#pragma unroll
  for (int a = 0; a < 16; ++a) {
    xv[a] = xs[wave][a][lane];  // conflict-free: consecutive lanes -> consecutive banks
    yv[a] = ys[wave][a][lane];
  }

  float* outp = out + ((size_t)edge * 99) * 64 + (size_t)(half * 32 + lane);

  // 23 instructions, output-row offsets are the running concat offsets (total 99).
  do_ins<0, 0, 0, 0>(xv, yv, outp);
  do_ins<0, 1, 1, 1>(xv, yv, outp);
  do_ins<0, 2, 2, 4>(xv, yv, outp);
  do_ins<0, 3, 3, 9>(xv, yv, outp);
  do_ins<1, 0, 1, 16>(xv, yv, outp);
  do_ins<1, 1, 0, 19>(xv, yv, outp);
  do_ins<1, 1, 2, 20>(xv, yv, outp);
  do_ins<1, 2, 1, 25>(xv, yv, outp);
  do_ins<1, 2, 3, 28>(xv, yv, outp);
  do_ins<1, 3, 2, 35>(xv, yv, outp);
  do_ins<2, 0, 2, 40>(xv, yv, outp);
  do_ins<2, 1, 1, 45>(xv, yv, outp);
  do_ins<2, 1, 3, 48>(xv, yv, outp);
  do_ins<2, 2, 0, 55>(xv, yv, outp);
  do_ins<2, 2, 2, 56>(xv, yv, outp);
  do_ins<2, 3, 1, 61>(xv, yv, outp);
  do_ins<2, 3, 3, 64>(xv, yv, outp);
  do_ins<3, 0, 3, 71>(xv, yv, outp);
  do_ins<3, 1, 2, 78>(xv, yv, outp);
  do_ins<3, 2, 1, 83>(xv, yv, outp);
  do_ins<3, 2, 3, 86>(xv, yv, outp);
  do_ins<3, 3, 0, 93>(xv, yv, outp);
  do_ins<3, 3, 2, 94>(xv, yv, outp);
}

extern "C" void kernel_launch(void* const* d_in, const int* in_sizes, int n_in,
                              void* d_out, int out_size, void* d_ws, size_t ws_size,
                              hipStream_t stream) {
  (void)n_in; (void)out_size; (void)d_ws; (void)ws_size;
  const float* x = (const float*)d_in[0];
  const float* y = (const float*)d_in[1];
  float* out = (float*)d_out;
  const int n_edges = in_sizes[0] / (16 * 64);  // [N,16,64] f32
  const int blocks = (n_edges + 3) / 4;         // 4 edges per 256-thread block
  tp_kernel<<<blocks, 256, 0, stream>>>(x, y, out, n_edges);
}